// PointPillarScatter_9259949490554
// MI455X (gfx1250) — compile-verified
//
#include <hip/hip_runtime.h>
#include <hip/hip_bf16.h>

// PointPillarScatter for MI455X (gfx1250).
// Pure bandwidth op: 219MB zero-fill + 12MB sparse scatter. No FLOPs -> no WMMA.
// Strategy: b128 non-temporal streaming stores for the fill (don't thrash the
// 192MB L2), coalesced 128B/wave feature reads + scattered b32 stores for the
// scatter, global_prefetch_b8 to hide coord latency.

typedef __attribute__((ext_vector_type(4))) float f32x4;

// ---------------------------------------------------------------------------
// Kernel 1: zero-fill the dense grid with 16B non-temporal stores.
// Each thread writes 4 float4s (64B); consecutive lanes write consecutive 16B
// chunks within each of the 4 sub-passes -> fully coalesced 512B/wave stores.
// ---------------------------------------------------------------------------
__global__ void pps_zero_fill_v4(f32x4* __restrict__ out, long n4) {
    const long i0 = (long)blockIdx.x * (4L * blockDim.x) + threadIdx.x;
    const f32x4 z = {0.0f, 0.0f, 0.0f, 0.0f};
#pragma unroll
    for (int k = 0; k < 4; ++k) {
        long i = i0 + (long)k * blockDim.x;
        if (i < n4) {
            // global_store_b128 ... th:TH_NT — stream, don't pollute L2
            __builtin_nontemporal_store(z, out + i);
        }
    }
}

// Scalar tail (only launched if out_size % 4 != 0; with C=64 it is not).
__global__ void pps_zero_fill_tail(float* __restrict__ out, long start, long n) {
    long i = start + (long)blockIdx.x * blockDim.x + threadIdx.x;
    if (i < n) __builtin_nontemporal_store(0.0f, out + i);
}

// ---------------------------------------------------------------------------
// Kernel 2: scatter. One block per pillar, blockDim.x == C (64 -> 2 wave32s).
//   read : feat[p*C + c]          -> contiguous 128B per wave (coalesced)
//   write: out[(b*C + c)*G + cell] -> scattered b32 (inherent to layout)
// mask[p] <= 0 is a uniform (SALU) whole-block early exit.
// ---------------------------------------------------------------------------
__global__ void pps_scatter(const float* __restrict__ feat,
                            const int*   __restrict__ coords,
                            const int*   __restrict__ mask,
                            const int*   __restrict__ nx_p,
                            const int*   __restrict__ ny_p,
                            const int*   __restrict__ nz_p,
                            float*       __restrict__ out,
                            int P, int C) {
    const int p = blockIdx.x;
    if (p >= P) return;

    // CDNA5 prefetch of a later pillar's coords -> global_prefetch_b8
    if (p + 64 < P) {
        __builtin_prefetch(coords + 4L * (p + 64), 0, 1);
    }

    if (mask[p] <= 0) return;  // uniform: whole block retires early

    const int c = threadIdx.x;
    if (c >= C) return;

    const int b = coords[4L * p + 0];
    const int x = coords[4L * p + 1];
    const int y = coords[4L * p + 2];
    const int z = coords[4L * p + 3];

    const int nx = *nx_p;
    const int ny = *ny_p;
    const int nz = *nz_p;

    const long G    = (long)nx * (long)ny * (long)nz;   // cells per batch
    const long cell = (long)x + (long)y * (long)nx + (long)z;
    if (cell < 0 || cell >= G) return;                  // reference drops OOB

    const float v = feat[(long)p * C + c];              // coalesced load
    out[((long)b * C + c) * G + cell] = v;              // scattered b32 store
}

// ---------------------------------------------------------------------------
// Host launcher. Inputs (setup_inputs order):
//   0: pillar_features f32 [P, C]
//   1: coords          i32 [P, 4]
//   2: mask            i32 [P]
//   3: batch_size  4: nx  5: ny  6: nz  7: num_features   (device scalars)
// Output: f32 [B, C*nz, ny, nx] flat (out_size elements).
// ---------------------------------------------------------------------------
extern "C" void kernel_launch(void* const* d_in, const int* in_sizes, int n_in,
                              void* d_out, int out_size, void* d_ws, size_t ws_size,
                              hipStream_t stream) {
    const float* feat   = (const float*)d_in[0];
    const int*   coords = (const int*)  d_in[1];
    const int*   mask   = (const int*)  d_in[2];
    const int*   nx_p   = (const int*)  d_in[4];
    const int*   ny_p   = (const int*)  d_in[5];
    const int*   nz_p   = (const int*)  d_in[6];
    float*       out    = (float*)d_out;

    const int P = in_sizes[2];                       // mask has one entry per pillar
    const int C = (P > 0) ? (in_sizes[0] / P) : 1;   // channels

    // --- fill ---
    const long n4  = (long)out_size / 4;             // full float4s
    const long rem = (long)out_size - n4 * 4;
    if (n4 > 0) {
        const int  threads = 256;
        const long perBlk  = 4L * threads;           // float4s per block
        const long blocks  = (n4 + perBlk - 1) / perBlk;
        pps_zero_fill_v4<<<(unsigned)blocks, threads, 0, stream>>>((f32x4*)out, n4);
    }
    if (rem > 0) {
        pps_zero_fill_tail<<<1, 256, 0, stream>>>(out, n4 * 4, (long)out_size);
    }

    // --- scatter (same stream -> ordered after fill) ---
    if (P > 0) {
        const int threads = (C <= 1024) ? C : 1024;  // C==64 here -> 2 wave32s
        pps_scatter<<<P, threads, 0, stream>>>(feat, coords, mask,
                                               nx_p, ny_p, nz_p, out, P, C);
    }
}